// SIRRollForward_65481071395073
// MI455X (gfx1250) — compile-verified
//
#include <hip/hip_runtime.h>
#include <stdint.h>

// SIR epidemic rollout, B independent rows, sequential scan over H steps.
// Memory-bound (~1.28 GB traffic, ~55us floor at 23.3 TB/s). CDNA5 paths:
//   - GLOBAL_LOAD_ASYNC_TO_LDS_B128 staging of the rate arrays (ASYNCcnt)
//   - GLOBAL_STORE_ASYNC_FROM_LDS_B32 coalesced transposed trajectory stores
//   - global_prefetch of the next chunk, padded bank-conflict-free LDS tiles

#define BT  128   // batch rows per block (= threads per block, 4 waves)
#define HC  32    // time-steps per LDS chunk (32 floats = 128B = 1 cacheline/row)
#define HCP 33    // padded LDS row stride in floats (33 coprime with 64 banks)

#define DT_EPS  1e-8f
#define RES_CLIP 10000.0f

typedef float f4v __attribute__((ext_vector_type(4)));
typedef int   v4i __attribute__((vector_size(4 * sizeof(int))));
typedef __attribute__((address_space(1))) v4i* gbl_v4i_p;   // global
typedef __attribute__((address_space(3))) v4i* lds_v4i_p;   // LDS
typedef __attribute__((address_space(1))) int* gbl_i_p;
typedef __attribute__((address_space(3))) int* lds_i_p;

__device__ __forceinline__ void async_copy_f4(const float* __restrict__ gsrc,
                                              float* lds_dst) {
#if __has_builtin(__builtin_amdgcn_global_load_async_to_lds_b128)
  __builtin_amdgcn_global_load_async_to_lds_b128(
      (gbl_v4i_p)gsrc, (lds_v4i_p)lds_dst, /*imm offset*/0, /*cpol*/0);
#else
  *(f4v*)lds_dst = *(const f4v*)gsrc;   // fallback: plain load + ds_store
#endif
}

__device__ __forceinline__ void async_store_f1(float* gdst, const float* lds_src) {
#if __has_builtin(__builtin_amdgcn_global_store_async_from_lds_b32)
  __builtin_amdgcn_global_store_async_from_lds_b32(
      (gbl_i_p)gdst, (lds_i_p)lds_src, /*imm offset*/0, /*cpol*/0);
#else
  *gdst = *lds_src;                     // fallback: ds_load + global store
#endif
}

__device__ __forceinline__ void wait_async_zero() {
#if __has_builtin(__builtin_amdgcn_s_wait_asynccnt)
  __builtin_amdgcn_s_wait_asynccnt(0);
#else
  asm volatile("s_wait_asynccnt 0" ::: "memory");
#endif
}

__global__ __launch_bounds__(BT) void sir_roll_kernel(
    const float* __restrict__ beta,   // [B,H]
    const float* __restrict__ gamma,  // [B,H]
    const float* __restrict__ mu,     // [B,H]
    const float* __restrict__ S0v,    // [B]
    const float* __restrict__ I0v,    // [B]
    const float* __restrict__ R0v,    // [B]
    const float* __restrict__ pop,    // [B]
    float* __restrict__ out,
    int B, int H) {
  // Tiles double as input (beta/gamma/mu) and output (S/I/R) staging.
  __shared__ float tileA[BT * HCP];   // beta  -> S
  __shared__ float tileB[BT * HCP];   // gamma -> I
  __shared__ float tileC[BT * HCP];   // mu    -> R

  const int t     = threadIdx.x;
  const int bbase = blockIdx.x * BT;
  const int b     = bbase + t;
  const bool active = (b < B);
  const int Hp1 = H + 1;

  // output section offsets (floats), in tuple return order
  const size_t secT = (size_t)B * (size_t)Hp1;   // traj section size
  const size_t secH = (size_t)B * (size_t)H;     // per-step section size
  const size_t oS  = 0;
  const size_t oI  = secT;
  const size_t oR  = 2 * secT;
  const size_t oD  = 3 * secT;
  const size_t oDe = 4 * secT;                   // death_flow
  const size_t oRe = 4 * secT + secH;            // physics_residual
  const size_t oBe = 4 * secT + 2 * secH;        // beta_t echo

  // ---- initial state: normalize to population ----
  float N = 1.0f, S = 0.f, I = 0.f, R = 0.f, D = 0.f;
  if (active) {
    N = pop[b];
    const float s0 = S0v[b], i0 = I0v[b], r0 = R0v[b];
    const float sc0 = N / (s0 + i0 + r0 + DT_EPS);
    S = s0 * sc0; I = i0 * sc0; R = r0 * sc0; D = 0.f;
    const size_t row0 = (size_t)b * Hp1;
    out[oS + row0] = S;
    out[oI + row0] = I;
    out[oR + row0] = R;
    out[oD + row0] = D;
  }

  const bool vec_out = ((H & 3) == 0);
  const int scol  = t & (HC - 1);   // coalesced-store column for this lane
  const int srow0 = t >> 5;         // base row (rows advance by BT/HC per k)

  for (int h0 = 0; h0 < H; h0 += HC) {
    const int hrem = H - h0;
    const int hcur = (hrem < HC) ? hrem : HC;

    // ---- cooperative async load of [BT x HC] rate tiles ----
    // 8 lanes cover one row's 128B => full-cacheline fetches.
#pragma unroll
    for (int k = 0; k < HC / 4; ++k) {          // BT*HC/4 f4s / BT threads
      const int i  = t + k * BT;
      const int bl = i >> 3;                    // i / (HC/4)
      int hl = (i & ((HC / 4) - 1)) << 2;
      if (hl + 4 > hrem) hl = (hrem >= 4) ? (hrem - 4) : 0;  // OOB-safe clamp
      int gb = bbase + bl; if (gb >= B) gb = B - 1;
      const size_t g = (size_t)gb * H + (size_t)(h0 + hl);
      const int l = bl * HCP + hl;
      async_copy_f4(beta  + g, &tileA[l]);
      async_copy_f4(gamma + g, &tileB[l]);
      async_copy_f4(mu    + g, &tileC[l]);
    }
    if (active && (h0 + HC < H)) {              // prefetch next chunk into L2
      const size_t gp = (size_t)b * H + (size_t)(h0 + HC);
      __builtin_prefetch(beta  + gp, 0, 0);
      __builtin_prefetch(gamma + gp, 0, 0);
      __builtin_prefetch(mu    + gp, 0, 0);
    }
    wait_async_zero();
    __syncthreads();

    // ---- sequential recurrence; recycle tiles: rate[j] -> S/I/R[j] ----
    if (active) {
      float* la = &tileA[t * HCP];
      float* lb = &tileB[t * HCP];
      float* lc = &tileC[t * HCP];
      const size_t rowT = (size_t)b * Hp1 + (size_t)h0 + 1;  // traj col h+1
      const size_t rowH = (size_t)b * H + (size_t)h0;

      float de4[4], re4[4], be4[4];
#pragma unroll 4
      for (int j = 0; j < hcur; ++j) {
        const float be = la[j], ga = lb[j], m = lc[j];
        const float flowSI = be * S * I / N;
        const float rec = ga * I;
        const float dth = m * I;
        float Sn = S - flowSI;
        float In = I + (flowSI - rec - dth);
        float Rn = R + rec;
        float Dn = D + dth;
        Sn = fmaxf(Sn, 0.f); In = fmaxf(In, 0.f);
        Rn = fmaxf(Rn, 0.f); Dn = fmaxf(Dn, 0.f);
        const float sc = N / (Sn + In + Rn + Dn + DT_EPS);
        Sn *= sc; In *= sc; Rn *= sc; Dn *= sc;
        const float dIe = flowSI - rec - dth;
        float res = (In - I) - dIe;
        res = res * res;
        res = fminf(res, RES_CLIP);

        la[j] = Sn;                  // S -> tileA (beta[j] already consumed)
        lb[j] = In;                  // I -> tileB
        lc[j] = Rn;                  // R -> tileC
        out[oD + rowT + j] = Dn;     // D: direct scalar (no spare tile)

        const int q = j & 3;
        de4[q] = dth; re4[q] = res; be4[q] = be;
        if (vec_out) {
          if (q == 3) {              // packed 16B stores (aligned sections)
            const size_t p = rowH + (size_t)(j - 3);
            *(f4v*)&out[oDe + p] = *(f4v*)de4;
            *(f4v*)&out[oRe + p] = *(f4v*)re4;
            *(f4v*)&out[oBe + p] = *(f4v*)be4;
          }
        } else {
          out[oDe + rowH + j] = dth;
          out[oRe + rowH + j] = res;
          out[oBe + rowH + j] = be;
        }

        S = Sn; I = In; R = Rn; D = Dn;
      }
    }
    __syncthreads();   // all rows' S/I/R values resident in tiles

    // ---- cooperative coalesced (transposed) trajectory stores ----
    // 32 lanes cover one row's 32 contiguous floats: 1-2 cachelines per
    // wave-store instead of 32. Issued on the async LDS->global path.
    if (scol < hcur) {
#pragma unroll 4
      for (int k = 0; k < BT / (BT / HC); ++k) {     // HC iterations
        const int row = k * (BT / HC) + srow0;       // 0..BT-1
        const int gb = bbase + row;
        if (gb < B) {
          const size_t g = (size_t)gb * Hp1 + (size_t)(h0 + 1 + scol);
          const int l = row * HCP + scol;
          async_store_f1(&out[oS + g], &tileA[l]);
          async_store_f1(&out[oI + g], &tileB[l]);
          async_store_f1(&out[oR + g], &tileC[l]);
        }
      }
    }
    wait_async_zero();   // tiles must be free before next chunk's loads
    __syncthreads();
  }
}

extern "C" void kernel_launch(void* const* d_in, const int* in_sizes, int n_in,
                              void* d_out, int out_size, void* d_ws, size_t ws_size,
                              hipStream_t stream) {
  const float* beta  = (const float*)d_in[0];
  const float* gamma = (const float*)d_in[1];
  const float* mu    = (const float*)d_in[2];
  const float* S0v   = (const float*)d_in[3];
  const float* I0v   = (const float*)d_in[4];
  const float* R0v   = (const float*)d_in[5];
  const float* pop   = (const float*)d_in[6];
  float* out = (float*)d_out;

  const int B = in_sizes[3];
  const int H = (B > 0) ? (in_sizes[0] / B) : 0;
  if (B <= 0 || H <= 0) return;

  const int grid = (B + BT - 1) / BT;
  hipLaunchKernelGGL(sir_roll_kernel, dim3(grid), dim3(BT), 0, stream,
                     beta, gamma, mu, S0v, I0v, R0v, pop, out, B, H);
}